// ProteinContactMapDecoder_55336358643504
// MI455X (gfx1250) — compile-verified
//
#include <hip/hip_runtime.h>

typedef __attribute__((ext_vector_type(2))) float v2f;
typedef __attribute__((ext_vector_type(8))) float v8f;

#define DDIM   1280          // feature dim
#define LOUT   510           // output rows/cols (512 sliced [1:-1])
#define BK     64            // K chunk staged in LDS
#define BTM    64            // block output rows
#define BTN    32            // block output cols
#define XSTR   (BK + 4)      // LDS row stride in floats (68: 16B-aligned, bank-conflict-free)

// out[i,j,k] = sum_d prev[i+1,d] * prev[j+1,d] * Wp[d,k] + b[k]
// (symmetrization annihilates the antisymmetric diff term; Wd is dead)
__global__ __launch_bounds__(128)
void contact_gram_wmma_f32(const float* __restrict__ prev,
                           const float* __restrict__ W,     // (2*DDIM, 2); Wp = first 2*DDIM floats
                           const float* __restrict__ bvec,  // (2,)
                           float* __restrict__ out)         // (LOUT, LOUT, 2)
{
    __shared__ float Xs [BTM * XSTR];   // raw X rows
    __shared__ float Ys0[BTN * XSTR];   // Y rows pre-scaled by Wp[:,0]
    __shared__ float Ys1[BTN * XSTR];   // Y rows pre-scaled by Wp[:,1]
    __shared__ float Wpl[2 * DDIM];

    const int t  = threadIdx.x;
    const int I0 = blockIdx.x * BTM;
    const int J0 = blockIdx.y * BTN;

    // Cache Wp (W[:1280,:2] row-major = first 2560 floats) in LDS once.
    for (int idx = t; idx < 2 * DDIM; idx += 128) Wpl[idx] = W[idx];
    __syncthreads();

    const int lane    = t & 31;
    const int wave    = t >> 5;
    const int wr      = (wave >> 1) * 32;   // wave's row offset within block tile (0 or 32)
    const int wc      = (wave & 1) * 16;    // wave's col offset within block tile (0 or 16)
    const int halfsel = lane >> 4;          // 0: lanes 0-15, 1: lanes 16-31
    const int mrow    = lane & 15;          // M (for A) / N (for B) index
    const int kbase   = halfsel * 2;        // K sub-offset per ISA A/B layout

    v8f c00 = {};   // rows wr+0..15,  channel 0
    v8f c10 = {};   // rows wr+16..31, channel 0
    v8f c01 = {};   // rows wr+0..15,  channel 1
    v8f c11 = {};   // rows wr+16..31, channel 1

    for (int k0 = 0; k0 < DDIM; k0 += BK) {
        __syncthreads();   // previous chunk fully consumed before overwrite

        // Stage X: 64 rows x 16 float4 = 1024 slots; 128 threads x 8 iters.
        #pragma unroll
        for (int it = 0; it < 8; ++it) {
            int e   = it * 128 + t;
            int row = e >> 4;            // 0..63
            int c4  = (e & 15) << 2;     // 0,4,...,60
            int gi  = I0 + row;
            float4 v = make_float4(0.f, 0.f, 0.f, 0.f);
            if (gi < LOUT) v = *(const float4*)(prev + (size_t)(gi + 1) * DDIM + k0 + c4);
            *(float4*)(Xs + row * XSTR + c4) = v;
        }
        // Stage Y pre-scaled by both Wp channels: 32 rows x 16 float4 = 512 slots; 4 iters.
        #pragma unroll
        for (int it = 0; it < 4; ++it) {
            int e   = it * 128 + t;
            int row = e >> 4;            // 0..31
            int c4  = (e & 15) << 2;
            int gj  = J0 + row;
            float4 v = make_float4(0.f, 0.f, 0.f, 0.f);
            if (gj < LOUT) v = *(const float4*)(prev + (size_t)(gj + 1) * DDIM + k0 + c4);
            // Wp row-major (d,k): float4 at 2*(k0+c4) = {wp0[d],wp1[d],wp0[d+1],wp1[d+1]}
            float4 wA = *(const float4*)(Wpl + 2 * (k0 + c4));       // d = c4, c4+1
            float4 wB = *(const float4*)(Wpl + 2 * (k0 + c4) + 4);   // d = c4+2, c4+3
            float4 y0 = make_float4(v.x * wA.x, v.y * wA.z, v.z * wB.x, v.w * wB.z);
            float4 y1 = make_float4(v.x * wA.y, v.y * wA.w, v.z * wB.y, v.w * wB.w);
            *(float4*)(Ys0 + row * XSTR + c4) = y0;
            *(float4*)(Ys1 + row * XSTR + c4) = y1;
        }
        __syncthreads();

        const float* xrow0 = Xs  + (wr + mrow) * XSTR;        // A tile 0: rows wr+0..15
        const float* xrow1 = Xs  + (wr + 16 + mrow) * XSTR;   // A tile 1: rows wr+16..31
        const float* yrow0 = Ys0 + (wc + mrow) * XSTR;        // B, channel 0
        const float* yrow1 = Ys1 + (wc + mrow) * XSTR;        // B, channel 1

        // Pure WMMA hot loop: 4 b64 LDS loads -> 4 WMMAs per k-step, no VALU.
        #pragma unroll
        for (int kk = 0; kk < BK; kk += 4) {
            const int d = kk + kbase;                // this lane's K pair start
            v2f a0 = *(const v2f*)(xrow0 + d);
            v2f a1 = *(const v2f*)(xrow1 + d);
            v2f b0 = *(const v2f*)(yrow0 + d);
            v2f b1 = *(const v2f*)(yrow1 + d);
            c00 = __builtin_amdgcn_wmma_f32_16x16x4_f32(false, a0, false, b0,
                                                        (short)0, c00, false, false);
            c10 = __builtin_amdgcn_wmma_f32_16x16x4_f32(false, a1, false, b0,
                                                        (short)0, c10, false, false);
            c01 = __builtin_amdgcn_wmma_f32_16x16x4_f32(false, a0, false, b1,
                                                        (short)0, c01, false, false);
            c11 = __builtin_amdgcn_wmma_f32_16x16x4_f32(false, a1, false, b1,
                                                        (short)0, c11, false, false);
        }
    }

    const float bias0 = bvec[0];
    const float bias1 = bvec[1];

    // C/D layout: VGPR r holds M = r + 8*halfsel, N = lane&15.
    const int j = J0 + wc + mrow;
    #pragma unroll
    for (int r = 0; r < 8; ++r) {
        int i0 = I0 + wr + r + 8 * halfsel;
        int i1 = i0 + 16;
        if (j < LOUT) {
            if (i0 < LOUT) {
                float2 o = make_float2(c00[r] + bias0, c01[r] + bias1);
                *(float2*)(out + ((size_t)i0 * LOUT + j) * 2) = o;
            }
            if (i1 < LOUT) {
                float2 o = make_float2(c10[r] + bias0, c11[r] + bias1);
                *(float2*)(out + ((size_t)i1 * LOUT + j) * 2) = o;
            }
        }
    }
}

extern "C" void kernel_launch(void* const* d_in, const int* in_sizes, int n_in,
                              void* d_out, int out_size, void* d_ws, size_t ws_size,
                              hipStream_t stream) {
    const float* prev = (const float*)d_in[0];   // (1, 512, 1280) f32
    const float* W    = (const float*)d_in[1];   // (2560, 2) f32
    const float* bv   = (const float*)d_in[2];   // (2,) f32
    float* out        = (float*)d_out;           // (1, 510, 510, 2) f32

    dim3 grid((LOUT + BTM - 1) / BTM, (LOUT + BTN - 1) / BTN);   // 8 x 16
    contact_gram_wmma_f32<<<grid, dim3(128), 0, stream>>>(prev, W, bv, out);
}